// HealEncoding_987842478249
// MI455X (gfx1250) — compile-verified
//
#include <hip/hip_runtime.h>
#include <stdint.h>

// Problem constants (match reference).
#define N_LEVELS 8
#define F_DIM    2
#define BATCH    524288            // 2^19
#define N_PIX    (12 * 128 * 128)  // 196608
#define N_NEIGH  8
#define N_LB     (N_LEVELS * BATCH)
#define N_TOT    (N_LEVELS * N_NEIGH * BATCH)

__device__ __forceinline__ float great_circle(float lat1, float lon1,
                                              float lat2, float lon2) {
    float sdlat = __sinf(0.5f * (lat2 - lat1));
    float sdlon = __sinf(0.5f * (lon2 - lon1));
    float a  = fmaf(__cosf(lat1) * __cosf(lat2), sdlon * sdlon, sdlat * sdlat);
    float s1 = sqrtf(fmaxf(a, 0.0f));
    float s2 = sqrtf(fmaxf(1.0f - a, 0.0f));
    return 2.0f * atan2f(s1, s2);
}

__global__ void heal_init(unsigned* maxSlot) { *maxSlot = 0u; }

// Pass 1: one thread per (level, point). Computes 8 neighbor distances,
// optionally stores them to the workspace, and reduces a global max
// (float bits as u32 are order-preserving since dist >= 0).
template <bool STORE>
__global__ __launch_bounds__(1024)
void heal_dist_max(const float2* __restrict__ pix,
                   const float2* __restrict__ neigh,
                   float* __restrict__ dist,
                   unsigned* __restrict__ maxSlot) {
    const int tid = threadIdx.x;
    const int gid = blockIdx.x * 1024 + tid;   // 0 .. L*B-1, exact cover
    const int l   = gid >> 19;                 // BATCH = 2^19
    const int b   = gid & (BATCH - 1);

    const float2 p = pix[(size_t)l * BATCH + b];
    float mx = 0.0f;
#pragma unroll
    for (int e = 0; e < N_NEIGH; ++e) {
        const size_t off = (size_t)(l * N_NEIGH + e) * BATCH + b;
        const float2 q = neigh[off];
        const float d = great_circle(p.x, p.y, q.x, q.y);
        if (STORE) dist[off] = d;
        mx = fmaxf(mx, d);
    }
    // wave32 butterfly reduce
#pragma unroll
    for (int s = 16; s > 0; s >>= 1) mx = fmaxf(mx, __shfl_xor(mx, s, 32));

    __shared__ float smax[32];
    const int wave = tid >> 5, lane = tid & 31;
    if (lane == 0) smax[wave] = mx;
    __syncthreads();
    if (wave == 0) {
        float m = smax[lane];                  // 32 waves in a 1024 block
#pragma unroll
        for (int s = 16; s > 0; s >>= 1) m = fmaxf(m, __shfl_xor(m, s, 32));
        if (lane == 0) atomicMax(maxSlot, __float_as_uint(m));
    }
}

// Pass 2 (workspace variant): pure data movement + gather.
// dist/idx/mask are staged into LDS one LEVEL (8 elements/lane) at a time
// via CDNA5 async global->LDS loads, double-buffered. Each lane stages only
// its own slots, so per-wave s_wait_asynccnt suffices (no block barrier).
// The inner 8-iteration compute loop contains no asm, so the compiler can
// keep all 8 L2-resident param gathers of a level in flight simultaneously.
__global__ __launch_bounds__(256)
void heal_gather_ws(const float2* __restrict__ params,
                    const float*  __restrict__ dist,
                    const int*    __restrict__ idx,
                    const unsigned char* __restrict__ mask,
                    const unsigned* __restrict__ maxSlot,
                    float* __restrict__ out) {
    __shared__ float         distT[2][N_NEIGH][256];
    __shared__ int           idxT [2][N_NEIGH][256];
    __shared__ unsigned char maskT[2][N_NEIGH * 256];

    const int tid = threadIdx.x;
    const int b   = blockIdx.x * 256 + tid;
    const float maxd = __uint_as_float(*maxSlot);

    float accx[N_LEVELS], accy[N_LEVELS];
#pragma unroll
    for (int l = 0; l < N_LEVELS; ++l) { accx[l] = 0.0f; accy[l] = 0.0f; }

    // Issue one level's 24 async loads (8x dist b32, 8x idx b32, 8x mask b8).
    auto issue_level = [&](int l, int buf) {
#pragma unroll
        for (int e = 0; e < N_NEIGH; ++e) {
            const int k   = l * N_NEIGH + e;
            const int go4 = 4 * (k * BATCH + b);     // < 2^28, fits i32
            const int go1 = k * BATCH + b;
            const unsigned ld = (unsigned)(uintptr_t)&distT[buf][e][tid];
            const unsigned li = (unsigned)(uintptr_t)&idxT [buf][e][tid];
            const unsigned lm = (unsigned)(uintptr_t)&maskT[buf][e * 256 + tid];
            asm volatile("global_load_async_to_lds_b32 %0, %1, %2"
                         :: "v"(ld), "v"(go4), "s"(dist) : "memory");
            asm volatile("global_load_async_to_lds_b32 %0, %1, %2"
                         :: "v"(li), "v"(go4), "s"(idx)  : "memory");
            asm volatile("global_load_async_to_lds_b8 %0, %1, %2"
                         :: "v"(lm), "v"(go1), "s"(mask) : "memory");
        }
    };

    issue_level(0, 0);
#pragma unroll
    for (int l = 0; l < N_LEVELS; ++l) {
        const int cur = l & 1;
        if (l + 1 < N_LEVELS) {
            // Buffer cur^1 was last read at level l-1: retire those DS reads,
            // prefetch level l+1 into it, then wait until only those 24 newest
            // async ops remain outstanding (=> level l resident in LDS).
            asm volatile("s_wait_dscnt 0x0" ::: "memory");
            issue_level(l + 1, cur ^ 1);
            asm volatile("s_wait_asynccnt 0x18" ::: "memory");
        } else {
            asm volatile("s_wait_asynccnt 0x0" ::: "memory");
        }
        // No asm below: 8 DS reads + 8 param gathers overlap freely.
#pragma unroll
        for (int e = 0; e < N_NEIGH; ++e) {
            const float d = distT[cur][e][tid];
            const int   i = idxT [cur][e][tid];
            const float m = (float)maskT[cur][e * 256 + tid];
            const float w = (maxd - d) * m;
            const float2 pr = params[(size_t)l * N_PIX + i];  // L2-resident
            accx[l] = fmaf(w, pr.x, accx[l]);
            accy[l] = fmaf(w, pr.y, accy[l]);
        }
    }

    // out[b, f*8 + l]: 64 contiguous bytes per thread.
    float4* o4 = (float4*)(out + (size_t)b * 16);
    o4[0] = make_float4(accx[0], accx[1], accx[2], accx[3]);
    o4[1] = make_float4(accx[4], accx[5], accx[6], accx[7]);
    o4[2] = make_float4(accy[0], accy[1], accy[2], accy[3]);
    o4[3] = make_float4(accy[4], accy[5], accy[6], accy[7]);
}

// Pass 2 fallback (workspace too small): recompute distances.
__global__ __launch_bounds__(256)
void heal_gather_rc(const float2* __restrict__ params,
                    const float2* __restrict__ pix,
                    const float2* __restrict__ neigh,
                    const int*    __restrict__ idx,
                    const unsigned char* __restrict__ mask,
                    const unsigned* __restrict__ maxSlot,
                    float* __restrict__ out) {
    const int b = blockIdx.x * 256 + threadIdx.x;
    const float maxd = __uint_as_float(*maxSlot);

    float accx[N_LEVELS], accy[N_LEVELS];
#pragma unroll
    for (int l = 0; l < N_LEVELS; ++l) { accx[l] = 0.0f; accy[l] = 0.0f; }

#pragma unroll
    for (int l = 0; l < N_LEVELS; ++l) {
        const float2 p = pix[(size_t)l * BATCH + b];
#pragma unroll
        for (int e = 0; e < N_NEIGH; ++e) {
            const size_t off = (size_t)(l * N_NEIGH + e) * BATCH + b;
            const float2 q = neigh[off];
            const float d = great_circle(p.x, p.y, q.x, q.y);
            const float m = (float)mask[off];
            const float w = (maxd - d) * m;
            const float2 pr = params[(size_t)l * N_PIX + idx[off]];
            accx[l] = fmaf(w, pr.x, accx[l]);
            accy[l] = fmaf(w, pr.y, accy[l]);
        }
    }
    float4* o4 = (float4*)(out + (size_t)b * 16);
    o4[0] = make_float4(accx[0], accx[1], accx[2], accx[3]);
    o4[1] = make_float4(accx[4], accx[5], accx[6], accx[7]);
    o4[2] = make_float4(accy[0], accy[1], accy[2], accy[3]);
    o4[3] = make_float4(accy[4], accy[5], accy[6], accy[7]);
}

extern "C" void kernel_launch(void* const* d_in, const int* in_sizes, int n_in,
                              void* d_out, int out_size, void* d_ws, size_t ws_size,
                              hipStream_t stream) {
    const float2*        params = (const float2*)d_in[0];
    const float2*        pix    = (const float2*)d_in[1];
    const float2*        neigh  = (const float2*)d_in[2];
    const int*           nidx   = (const int*)d_in[3];
    const unsigned char* nmask  = (const unsigned char*)d_in[4];  // jax bool = 1 byte
    float*               out    = (float*)d_out;

    unsigned* maxSlot = (unsigned*)d_ws;
    float*    distBuf = (float*)((char*)d_ws + 256);
    const size_t need = 256 + (size_t)N_TOT * sizeof(float);
    const bool useWs  = (ws_size >= need);

    heal_init<<<1, 1, 0, stream>>>(maxSlot);

    const dim3 g1(N_LB / 1024), b1(1024);
    const dim3 g2(BATCH / 256), b2(256);
    if (useWs) {
        heal_dist_max<true ><<<g1, b1, 0, stream>>>(pix, neigh, distBuf, maxSlot);
        heal_gather_ws<<<g2, b2, 0, stream>>>(params, distBuf, nidx, nmask, maxSlot, out);
    } else {
        heal_dist_max<false><<<g1, b1, 0, stream>>>(pix, neigh, nullptr, maxSlot);
        heal_gather_rc<<<g2, b2, 0, stream>>>(params, pix, neigh, nidx, nmask, maxSlot, out);
    }
}